// BILTDetector_29978871726228
// MI455X (gfx1250) — compile-verified
//
#include <hip/hip_runtime.h>
#include <stdint.h>

// ---------------- problem constants (match reference) ----------------
#define BATCH   16
#define NA      16320      // anchors
#define NC      80         // classes
#define NSORT   16384      // anchors padded to pow2 for bitonic sort
#define KPAD    1024       // K_PRE=1000 padded to pow2
#define KPRE    1000
#define MAXDET  300
#define IMGSZ   512.0f
#define SCORE_THR 0.05f
#define IOU_THR 0.5f
#define CLS_OFF 10000.0f

typedef float v2f __attribute__((ext_vector_type(2)));
typedef float v8f __attribute__((ext_vector_type(8)));
typedef int   v4i __attribute__((ext_vector_type(4)));

// Async global->LDS (gfx1250): guarded so either toolchain (and the host
// pass) still compiles; when available it emits global_load_async_to_lds_b128
// tracked by ASYNCcnt + s_wait_asynccnt.  Probe round 2 showed the builtin's
// param type is (AS1 v4i*, AS3 v4i*, imm offset, imm cpol).
#if defined(__AMDGCN__) && __has_builtin(__builtin_amdgcn_global_load_async_to_lds_b128) && \
    __has_builtin(__builtin_amdgcn_s_wait_asynccnt)
#define HAS_ASYNC_LDS 1
typedef __attribute__((address_space(1))) v4i* gv4i_ptr;   // global
typedef __attribute__((address_space(3))) v4i* lv4i_ptr;   // LDS
__device__ inline void async_g2l_b128(const void* g, void* l) {
  __builtin_amdgcn_global_load_async_to_lds_b128(
      (gv4i_ptr)(unsigned long long)(uintptr_t)g,   // inttoptr -> AS1
      (lv4i_ptr)l,                                  // addrspacecast -> AS3
      /*offset=*/0, /*cpol=*/0);
}
#else
#define HAS_ASYNC_LDS 0
#endif

// =====================================================================
// Kernel 1: per-anchor class max (sigmoid is monotonic -> max on logits,
// one sigmoid at the end) + box decode. This kernel carries ~84MB of the
// ~88MB total traffic => it IS the roofline; everything is float4 loads.
// =====================================================================
__global__ __launch_bounds__(256)
void k_score_decode(const float* __restrict__ cls,
                    const float* __restrict__ boxp,
                    const float* __restrict__ anchors,
                    float* __restrict__ sc, int* __restrict__ lb,
                    float* __restrict__ bx) {
  int a = blockIdx.x * blockDim.x + threadIdx.x;
  int b = blockIdx.y;
  if (a >= NA) return;
  const float4* cp = (const float4*)(cls + ((size_t)b * NA + a) * NC);
  float m = -1e30f; int mi = 0;
  #pragma unroll
  for (int q = 0; q < NC / 4; ++q) {
    __builtin_prefetch((const void*)(cp + q + 8), 0, 0);   // global_prefetch_b8
    float4 v = cp[q];
    if (v.x > m) { m = v.x; mi = 4 * q + 0; }
    if (v.y > m) { m = v.y; mi = 4 * q + 1; }
    if (v.z > m) { m = v.z; mi = 4 * q + 2; }
    if (v.w > m) { m = v.w; mi = 4 * q + 3; }
  }
  float s = 1.0f / (1.0f + __expf(-m));
  sc[(size_t)b * NA + a] = (s > SCORE_THR) ? s : -1.0f;
  lb[(size_t)b * NA + a] = mi;

  float4 an = ((const float4*)anchors)[a];
  float4 d  = ((const float4*)boxp)[(size_t)b * NA + a];
  float aw = an.z - an.x, ah = an.w - an.y;
  float acx = an.x + 0.5f * aw, acy = an.y + 0.5f * ah;
  float cx = d.x * aw + acx, cy = d.y * ah + acy;
  float w = __expf(d.z) * aw, h = __expf(d.w) * ah;
  float4 o;
  o.x = fminf(fmaxf(cx - 0.5f * w, 0.f), IMGSZ);
  o.y = fminf(fmaxf(cy - 0.5f * h, 0.f), IMGSZ);
  o.z = fminf(fmaxf(cx + 0.5f * w, 0.f), IMGSZ);
  o.w = fminf(fmaxf(cy + 0.5f * h, 0.f), IMGSZ);
  ((float4*)bx)[(size_t)b * NA + a] = o;
}

// =====================================================================
// Kernel 2: full in-LDS bitonic sort of 16384 (score,idx) keys per image.
// CDNA5 WGP has 320KB LDS -> 128KB of keys fits in ONE workgroup; one
// block per image, 1024 threads. Then gather the top-1024 candidates
// (score, label, output box, class-offset box, area).
// =====================================================================
__global__ __launch_bounds__(1024)
void k_topk_sort(const float* __restrict__ sc, const int* __restrict__ lb,
                 const float* __restrict__ bx,
                 float* __restrict__ cscore, int* __restrict__ clabel,
                 float* __restrict__ cobox, float* __restrict__ cbout,
                 float* __restrict__ carea) {
  extern __shared__ unsigned long long keys[];   // 16384 * 8B = 128KB
  int b = blockIdx.x;
  int t = threadIdx.x;
  for (int i = t; i < NSORT; i += 1024) {
    unsigned long long key = 0ull;                // padding = smallest
    if (i < NA) {
      unsigned u = __float_as_uint(sc[(size_t)b * NA + i]);
      u = (u & 0x80000000u) ? ~u : (u | 0x80000000u);   // order-preserving
      key = ((unsigned long long)u << 32) | (unsigned)i;
    }
    keys[i] = key;
  }
  __syncthreads();
  for (unsigned k = 2; k <= NSORT; k <<= 1) {
    for (unsigned j = k >> 1; j > 0; j >>= 1) {
      for (unsigned i = t; i < NSORT; i += 1024) {
        unsigned ixj = i ^ j;
        if (ixj > i) {
          unsigned long long x = keys[i], y = keys[ixj];
          bool descBlock = ((i & k) == 0);        // overall descending
          if (descBlock ? (x < y) : (x > y)) { keys[i] = y; keys[ixj] = x; }
        }
      }
      __syncthreads();
    }
  }
  if (t < KPAD) {
    int idx = (int)(keys[t] & 0xFFFFFFFFull);     // top-1024 are all real
    float s = sc[(size_t)b * NA + idx];
    if (t >= KPRE) s = -1.0f;                     // reference keeps only 1000
    cscore[b * KPAD + t] = s;
    int l = lb[(size_t)b * NA + idx];
    clabel[b * KPAD + t] = l;
    float4 bb = ((const float4*)bx)[(size_t)b * NA + idx];
    ((float4*)cbout)[b * KPAD + t] = bb;          // un-offset box for output
    float off = (float)l * CLS_OFF;               // per-class NMS offset
    float4 ob = make_float4(bb.x + off, bb.y + off, bb.z + off, bb.w + off);
    ((float4*)cobox)[b * KPAD + t] = ob;
    carea[b * KPAD + t] = fmaxf(bb.z - bb.x, 0.f) * fmaxf(bb.w - bb.y, 0.f);
  }
}

// =====================================================================
// Kernel 3: 1024x1024 suppression bitmask per image, 16x16 tiles/wave.
// Union base (area_i + area_j) is a rank-2 matmul -> V_WMMA_F32_16X16X4_F32
// on the matrix pipe co-executing with the VALU min/max intersection math.
// iou > thr tested as inter > thr*union (division-free).  WMMA C layout
// (VGPR k: lanes0-15 row M=k, lanes16-31 row M=k+8) means one ballot_w32
// yields two 16-bit mask words; lanes 0-15 then store the 16 words of the
// tile in parallel (single divergent region instead of 8).
// =====================================================================
__global__ __launch_bounds__(256)
void k_iou_mask(const float* __restrict__ carea, const float* __restrict__ cobox,
                unsigned short* __restrict__ mask) {
  int b    = blockIdx.y;
  int wave = threadIdx.x >> 5;
  int lane = threadIdx.x & 31;
  int tile = blockIdx.x * 8 + wave;        // 64*64 = 4096 tiles / image
  int ti = tile >> 6, tj = tile & 63;
  int I = ti * 16, J = tj * 16;
  const float*  area = carea + b * KPAD;
  const float4* obox = (const float4*)cobox + b * KPAD;

  int  lo = lane & 15;
  bool fh = lane < 16;                     // first lane-half
  // A (16x4): rows = [area_m, 1, 0, 0] ; B (4x16): rows = [1; area_n; 0; 0]
  v2f av, bv;
  av.x = fh ? area[I + lo] : 0.0f;         // K=0 | K=2
  av.y = fh ? 1.0f         : 0.0f;         // K=1 | K=3
  bv.x = fh ? 1.0f         : 0.0f;
  bv.y = fh ? area[J + lo] : 0.0f;
  v8f sum = {};
  sum = __builtin_amdgcn_wmma_f32_16x16x4_f32(
      /*neg_a=*/false, av, /*neg_b=*/false, bv,
      /*c_mod=*/(short)0, sum, /*reuse_a=*/false, /*reuse_b=*/false);

  float4 cb = obox[J + lo];                // column box (N = lane&15)
  int mrow0 = I + (fh ? 0 : 8);
  unsigned bal[8];
  #pragma unroll
  for (int k = 0; k < 8; ++k) {
    float4 rb = obox[mrow0 + k];           // row box (L0-resident, reused 4096x)
    float lx = fmaxf(rb.x, cb.x), ly = fmaxf(rb.y, cb.y);
    float hx = fminf(rb.z, cb.z), hy = fminf(rb.w, cb.w);
    float inter = fmaxf(hx - lx, 0.f) * fmaxf(hy - ly, 0.f);
    float uni = fmaxf(sum[k] - inter, 1e-6f);
    bal[k] = __builtin_amdgcn_ballot_w32(inter > IOU_THR * uni);  // uniform
  }
  if (lane < 16) {
    int k = lane & 7;                      // select ballot (small cndmask tree)
    unsigned bw = bal[0];
    #pragma unroll
    for (int q = 1; q < 8; ++q) if (k == q) bw = bal[q];
    unsigned short val = (lane < 8) ? (unsigned short)(bw & 0xFFFFu)
                                    : (unsigned short)(bw >> 16);
    mask[((size_t)b * KPAD + (I + lane)) * 64 + tj] = val;   // row I+lane
  }
}

// =====================================================================
// Kernel 4: stage 128KB mask into LDS (async DMA when available), one
// wave32 runs the sequential greedy scan (lane l owns keep word l; the
// keep bit is broadcast via shfl; row i+1 is prefetched from LDS while
// row i is applied, so the serial chain is just shfl+and).  Candidates
// are score-sorted, so final top-300 == bit compaction.
// =====================================================================
__global__ __launch_bounds__(256)
void k_nms_scan(const unsigned short* __restrict__ mask,
                const float* __restrict__ cscore, const int* __restrict__ clabel,
                const float* __restrict__ cbout,
                float* __restrict__ out_box, float* __restrict__ out_sc,
                float* __restrict__ out_lb) {
  extern __shared__ unsigned maskLDS[];    // 1024 rows * 32 words = 128KB
  int b = blockIdx.x;
  int t = threadIdx.x;
  const uint4* gm = (const uint4*)(mask + (size_t)b * KPAD * 64);
  uint4* lm = (uint4*)maskLDS;
#if HAS_ASYNC_LDS
  for (int i = t; i < KPAD * 32 / 4; i += 256)
    async_g2l_b128((const void*)(gm + i), (void*)(lm + i));
  __builtin_amdgcn_s_wait_asynccnt(0);
#else
  for (int i = t; i < KPAD * 32 / 4; i += 256) lm[i] = gm[i];
#endif
  for (int i = t; i < MAXDET; i += 256) {  // zero outputs (padding entries)
    out_sc[b * MAXDET + i] = 0.f;
    out_lb[b * MAXDET + i] = 0.f;
    ((float4*)out_box)[b * MAXDET + i] = make_float4(0, 0, 0, 0);
  }
  __syncthreads();
  if (t < 32) {
    int lane = t;
    unsigned keepw = 0;                    // initial keep = score > 0
    for (int bit = 0; bit < 32; ++bit)
      if (cscore[b * KPAD + lane * 32 + bit] > 0.f) keepw |= 1u << bit;
    unsigned mwNext = maskLDS[lane];       // prefetch row 0
    for (int i = 0; i < KPAD; ++i) {       // greedy scan (inherently serial)
      unsigned mw = mwNext;
      if (i + 1 < KPAD) mwNext = maskLDS[(i + 1) * 32 + lane];  // pipeline
      int sl = i >> 5;
      unsigned kw = __shfl(keepw, sl, 32);
      if ((kw >> (i & 31)) & 1u) {
        unsigned filt;                     // only suppress j > i
        if (lane > sl)       filt = 0xFFFFFFFFu;
        else if (lane == sl) filt = ((i & 31) == 31) ? 0u
                                   : (0xFFFFFFFFu << ((i & 31) + 1));
        else                 filt = 0u;
        keepw &= ~(mw & filt);
      }
    }
    // exclusive prefix of popcounts across lanes
    unsigned pc = (unsigned)__popc(keepw), pre = 0;
    for (int l = 0; l < 32; ++l) {
      unsigned v = __shfl(pc, l, 32);
      if (l < lane) pre += v;
    }
    unsigned rem = keepw;
    while (rem) {
      int bit = __builtin_ctz(rem);
      rem &= rem - 1;
      unsigned below = (bit == 0) ? 0u : (keepw & ((1u << bit) - 1u));
      unsigned rank = pre + (unsigned)__popc(below);
      if (rank < MAXDET) {
        int i = lane * 32 + bit;
        out_sc[b * MAXDET + rank] = cscore[b * KPAD + i];
        out_lb[b * MAXDET + rank] = (float)(clabel[b * KPAD + i] + 1);
        ((float4*)out_box)[b * MAXDET + rank] = ((const float4*)cbout)[b * KPAD + i];
      }
    }
  }
}

// =====================================================================
extern "C" void kernel_launch(void* const* d_in, const int* in_sizes, int n_in,
                              void* d_out, int out_size, void* d_ws, size_t ws_size,
                              hipStream_t stream) {
  const float* cls     = (const float*)d_in[0];  // (16,16320,80)
  const float* boxp    = (const float*)d_in[1];  // (16,16320,4)
  const float* anchors = (const float*)d_in[2];  // (16320,4)

  // workspace carve-out (~9MB, 256B-aligned slices)
  char* ws = (char*)d_ws;
  size_t off = 0;
  auto take = [&](size_t bytes) -> void* {
    void* p = ws + off;
    off = (off + bytes + 255) & ~((size_t)255);
    return p;
  };
  float* sc     = (float*)take((size_t)BATCH * NA * 4);
  int*   lb     = (int*)  take((size_t)BATCH * NA * 4);
  float* bx     = (float*)take((size_t)BATCH * NA * 16);
  float* cscore = (float*)take((size_t)BATCH * KPAD * 4);
  int*   clabel = (int*)  take((size_t)BATCH * KPAD * 4);
  float* cobox  = (float*)take((size_t)BATCH * KPAD * 16);
  float* cbout  = (float*)take((size_t)BATCH * KPAD * 16);
  float* carea  = (float*)take((size_t)BATCH * KPAD * 4);
  unsigned short* mask = (unsigned short*)take((size_t)BATCH * KPAD * 64 * 2);

  // outputs concatenated flat: boxes(16,300,4) | scores(16,300) | labels(16,300)
  float* out_box = (float*)d_out;
  float* out_sc  = out_box + (size_t)BATCH * MAXDET * 4;
  float* out_lb  = out_sc  + (size_t)BATCH * MAXDET;

  k_score_decode<<<dim3((NA + 255) / 256, BATCH), 256, 0, stream>>>(
      cls, boxp, anchors, sc, lb, bx);
  k_topk_sort<<<BATCH, 1024, NSORT * sizeof(unsigned long long), stream>>>(
      sc, lb, bx, cscore, clabel, cobox, cbout, carea);
  k_iou_mask<<<dim3(512, BATCH), 256, 0, stream>>>(carea, cobox, mask);
  k_nms_scan<<<BATCH, 256, KPAD * 32 * sizeof(unsigned), stream>>>(
      mask, cscore, clabel, cbout, out_box, out_sc, out_lb);
}